// EnhancedESA3DEncoder_43499428774419
// MI455X (gfx1250) — compile-verified
//
#include <hip/hip_runtime.h>
#include <hip/hip_bf16.h>

typedef float v2f __attribute__((ext_vector_type(2)));
typedef float v8f __attribute__((ext_vector_type(8)));

#define E_EDGES 3072
#define EW      (E_EDGES / 32)   // mask words per row
#define D_MODEL 128
#define NHEAD   8
#define HDIM    16
#define RBF_R   64
#define FFN_D   512
#define CUTOFF_F 10.0f

// ---------------------------------------------------------------------------
// Exact-fp32 WMMA: D(16x16,f32) = A(16x4,f32) * B(4x16,f32) + C
// A layout: lanes 0-15 hold M=lane {K=k0,k0+1}; lanes 16-31 hold M=lane-16 {K=k0+2,k0+3}
// B layout: V0: lanes0-15 K=k0 / lanes16-31 K=k0+2 ; V1: K=k0+1 / K=k0+3 ; N=lane&15
// C/D layout: vgpr r, lanes0-15 -> M=r ; lanes16-31 -> M=r+8 ; N=lane&15
// ---------------------------------------------------------------------------
__device__ __forceinline__ v8f wmma4(v2f a, v2f b, v8f c) {
  return __builtin_amdgcn_wmma_f32_16x16x4_f32(
      /*neg_a=*/false, a, /*neg_b=*/false, b,
      /*c_mod=*/(short)0, c, /*reuse_a=*/false, /*reuse_b=*/false);
}

// ---------------------------------------------------------------------------
// utility kernels
// ---------------------------------------------------------------------------
__global__ __launch_bounds__(256) void copy_f32(const float* __restrict__ in,
                                                float* __restrict__ out, int n) {
  int i = blockIdx.x * blockDim.x + threadIdx.x;
  if (i < n) out[i] = in[i];
}

__global__ __launch_bounds__(256) void zero_f32(float* __restrict__ p, int n) {
  int i = blockIdx.x * blockDim.x + threadIdx.x;
  if (i < n) p[i] = 0.0f;
}

// RBF embedding of edge length with cosine cutoff: rbf[E, 64]
__global__ __launch_bounds__(256) void rbf_kernel(const int* __restrict__ ei,
                                                  const float* __restrict__ nc,
                                                  float* __restrict__ rbf) {
  int e = blockIdx.x * blockDim.x + threadIdx.x;
  if (e >= E_EDGES) return;
  int s = ei[e], d = ei[E_EDGES + e];
  float dx = nc[d * 3 + 0] - nc[s * 3 + 0];
  float dy = nc[d * 3 + 1] - nc[s * 3 + 1];
  float dz = nc[d * 3 + 2] - nc[s * 3 + 2];
  float dist = sqrtf(dx * dx + dy * dy + dz * dz);
  float t = dist * (1.0f / CUTOFF_F);
  t = fminf(fmaxf(t, 0.0f), 1.0f);
  float fc = 0.5f * (cosf(3.14159265358979323846f * t) + 1.0f);
  const float invw = (float)RBF_R / CUTOFF_F;  // 1/width = R/cutoff
  #pragma unroll 4
  for (int r = 0; r < RBF_R; ++r) {
    float c = CUTOFF_F * (float)r / (float)(RBF_R - 1);
    float u = (dist - c) * invw;
    rbf[e * RBF_R + r] = __expf(-u * u) * fc;
  }
}

// intra/inter masks as bitfields [E][E/32]
__global__ __launch_bounds__(256) void mask_kernel(const int* __restrict__ ei,
                                                   const int* __restrict__ bid,
                                                   unsigned* __restrict__ intraM,
                                                   unsigned* __restrict__ interM) {
  int idx = blockIdx.x * blockDim.x + threadIdx.x;  // over E*EW words
  if (idx >= E_EDGES * EW) return;
  int i = idx / EW, w = idx % EW;
  int si = ei[i], di = ei[E_EDGES + i];
  int bsi = bid[si], bdi = bid[di];
  unsigned mi = 0u, me = 0u;
  #pragma unroll 4
  for (int b = 0; b < 32; ++b) {
    int j = w * 32 + b;
    int sj = ei[j], dj = ei[E_EDGES + j];
    int bsj = bid[sj], bdj = bid[dj];
    bool intra = (bsi == bsj) && (bdi == bdj);
    bool share = (si == sj) || (si == dj) || (di == sj) || (di == dj);
    bool inter = share && !intra;
    if (i == j) { intra = true; inter = true; }
    mi |= (unsigned)intra << b;
    me |= (unsigned)inter << b;
  }
  intraM[idx] = mi;
  interM[idx] = me;
}

// ---------------------------------------------------------------------------
// Generic WMMA f32 GEMM: C[M,N] = act(A[M,K] @ B[K,N] (+bias) (+add))
// One wave computes a 16x32 output tile (two 16x16 accumulators), so each
// A fragment feeds two WMMAs and consecutive waves share A rows (cache bcast).
// Requires N % 32 == 0, K % 4 == 0. ACT: 0 none, 1 SiLU.
// ---------------------------------------------------------------------------
template <int ACT, bool ADD, bool BIAS>
__global__ __launch_bounds__(256) void gemm_wmma(const float* __restrict__ A,
                                                 const float* __restrict__ B,
                                                 const float* __restrict__ bias,
                                                 const float* __restrict__ addsrc,
                                                 float* __restrict__ C,
                                                 int M, int N, int K) {
  const int lane = threadIdx.x & 31;
  const int half = lane >> 4;
  const int l16  = lane & 15;
  const int wid  = (blockIdx.x * blockDim.x + threadIdx.x) >> 5;
  const int ncg  = N >> 5;                 // 32-col groups
  const int mi = wid / ncg, cg = wid - mi * ncg;
  if (mi >= (M >> 4)) return;
  const float* Ar = A + (size_t)((mi << 4) + l16) * K;
  const int col0 = (cg << 5) + l16;        // first 16-col tile
  const int col1 = col0 + 16;              // second 16-col tile
  v8f acc0 = {0.f, 0.f, 0.f, 0.f, 0.f, 0.f, 0.f, 0.f};
  v8f acc1 = {0.f, 0.f, 0.f, 0.f, 0.f, 0.f, 0.f, 0.f};
  for (int k = 0; k < K; k += 4) {
    const int ka = k + half * 2;
    v2f a; a.x = Ar[ka]; a.y = Ar[ka + 1];
    const float* B0 = B + (size_t)ka * N;
    const float* B1 = B + (size_t)(ka + 1) * N;
    v2f b0; b0.x = B0[col0]; b0.y = B1[col0];
    v2f b1; b1.x = B0[col1]; b1.y = B1[col1];
    acc0 = wmma4(a, b0, acc0);
    acc1 = wmma4(a, b1, acc1);
  }
  #pragma unroll
  for (int r = 0; r < 8; ++r) {
    const int orow = (mi << 4) + r + half * 8;
    const size_t o0 = (size_t)orow * N + col0;
    const size_t o1 = (size_t)orow * N + col1;
    float v0 = acc0[r], v1 = acc1[r];
    if (BIAS) { v0 += bias[col0]; v1 += bias[col1]; }
    if (ADD)  { v0 += addsrc[o0]; v1 += addsrc[o1]; }
    if (ACT == 1) {
      v0 = v0 / (1.0f + __expf(-v0));   // SiLU
      v1 = v1 / (1.0f + __expf(-v1));
    }
    C[o0] = v0;
    C[o1] = v1;
  }
}

// ---------------------------------------------------------------------------
// LayerNorm over D=128, one wave per row
// ---------------------------------------------------------------------------
__global__ __launch_bounds__(256) void layernorm_k(const float* __restrict__ in,
                                                   const float* __restrict__ g,
                                                   const float* __restrict__ b,
                                                   float* __restrict__ out) {
  const int lane = threadIdx.x & 31;
  const int row = (blockIdx.x * blockDim.x + threadIdx.x) >> 5;
  const float* p = in + (size_t)row * D_MODEL;
  float v[4], s = 0.f, s2 = 0.f;
  #pragma unroll
  for (int t = 0; t < 4; ++t) {
    v[t] = p[lane + 32 * t];
    s += v[t]; s2 += v[t] * v[t];
  }
  #pragma unroll
  for (int off = 16; off; off >>= 1) {
    s  += __shfl_xor(s, off, 32);
    s2 += __shfl_xor(s2, off, 32);
  }
  const float mean = s * (1.0f / D_MODEL);
  const float var  = s2 * (1.0f / D_MODEL) - mean * mean;
  const float inv  = rsqrtf(var + 1e-5f);
  float* q = out + (size_t)row * D_MODEL;
  #pragma unroll
  for (int t = 0; t < 4; ++t) {
    int c = lane + 32 * t;
    q[c] = (v[t] - mean) * inv * g[c] + b[c];
  }
}

// ---------------------------------------------------------------------------
// Flash attention over the masked edge set, one wave per (head, 16-row tile).
// INTRA additionally accumulates (mean_h P) @ x into px via atomics.
// ---------------------------------------------------------------------------
template <bool INTRA>
__global__ __launch_bounds__(256) void attn_flash(const float* __restrict__ q,
                                                  const float* __restrict__ k,
                                                  const float* __restrict__ v,
                                                  const unsigned* __restrict__ mask,
                                                  const float* __restrict__ xin,
                                                  float* __restrict__ out,
                                                  float* __restrict__ px) {
  __shared__ float pbuf[8 * 16 * 17];  // per-wave 16x16 P tile (padded)
  const int lane = threadIdx.x & 31;
  const int half = lane >> 4;
  const int l16  = lane & 15;
  const int wv   = threadIdx.x >> 5;
  float* P = pbuf + wv * (16 * 17);
  const int wid   = blockIdx.x * (blockDim.x >> 5) + wv;
  const int head  = wid & (NHEAD - 1);
  const int itile = wid >> 3;
  const int i0 = itile << 4;

  // Q fragments in A-layout, pre-scaled by 1/sqrt(hd)
  v2f qa[4];
  {
    const float* qp = q + (size_t)(i0 + l16) * D_MODEL + head * HDIM;
    #pragma unroll
    for (int s4 = 0; s4 < 4; ++s4) {
      const int kk = s4 * 4 + half * 2;
      qa[s4].x = qp[kk] * 0.25f;
      qa[s4].y = qp[kk + 1] * 0.25f;
    }
  }

  float m_[8], l_[8];
  #pragma unroll
  for (int r = 0; r < 8; ++r) { m_[r] = -1e30f; l_[r] = 0.f; }
  v8f o  = {0.f, 0.f, 0.f, 0.f, 0.f, 0.f, 0.f, 0.f};
  v8f xa = {0.f, 0.f, 0.f, 0.f, 0.f, 0.f, 0.f, 0.f};

  for (int j0 = 0; j0 < E_EDGES; j0 += 16) {
    // --- mask bits for this 16x16 tile; skip if entirely masked ---
    unsigned sel[8], anyb = 0;
    const int jw = j0 >> 5;
    const int sh = j0 & 16;
    #pragma unroll
    for (int r = 0; r < 8; ++r) {
      const int row = i0 + r + half * 8;
      sel[r] = (mask[row * EW + jw] >> sh) & 0xFFFFu;
      anyb |= sel[r];
    }
    if (!__any((int)(anyb != 0u))) continue;

    // --- S = (Q/sqrt(hd)) K^T via 4 WMMAs ---
    v8f sacc = {0.f, 0.f, 0.f, 0.f, 0.f, 0.f, 0.f, 0.f};
    const float* kp = k + (size_t)(j0 + l16) * D_MODEL + head * HDIM;
    #pragma unroll
    for (int s4 = 0; s4 < 4; ++s4) {
      const int kk = s4 * 4 + half * 2;
      v2f b; b.x = kp[kk]; b.y = kp[kk + 1];
      sacc = wmma4(qa[s4], b, sacc);
    }

    // --- masked online softmax (per-row state, rows split across lane halves) ---
    #pragma unroll
    for (int r = 0; r < 8; ++r) {
      float sv = ((sel[r] >> l16) & 1u) ? sacc[r] : -1e9f;
      float t = sv;
      t = fmaxf(t, __shfl_xor(t, 1, 16));
      t = fmaxf(t, __shfl_xor(t, 2, 16));
      t = fmaxf(t, __shfl_xor(t, 4, 16));
      t = fmaxf(t, __shfl_xor(t, 8, 16));
      const float mn = fmaxf(m_[r], t);
      const float scale = __expf(m_[r] - mn);
      const float pv = __expf(sv - mn);
      float rs = pv;
      rs += __shfl_xor(rs, 1, 16);
      rs += __shfl_xor(rs, 2, 16);
      rs += __shfl_xor(rs, 4, 16);
      rs += __shfl_xor(rs, 8, 16);
      l_[r] = l_[r] * scale + rs;
      m_[r] = mn;
      o[r] *= scale;
      if (INTRA) xa[r] *= scale;
      P[(r + half * 8) * 17 + l16] = pv;   // C-layout -> LDS
    }
    // make P visible across lanes of this wave (DS ops are in-order per wave)
    asm volatile("s_wait_dscnt 0x0" ::: "memory");

    // --- O += P @ V (and xa += P @ x for INTRA) via WMMA, P reloaded in A-layout ---
    #pragma unroll
    for (int s4 = 0; s4 < 4; ++s4) {
      const int kk = s4 * 4 + half * 2;
      v2f a; a.x = P[l16 * 17 + kk]; a.y = P[l16 * 17 + kk + 1];
      v2f b;
      b.x = v[(size_t)(j0 + kk) * D_MODEL + head * HDIM + l16];
      b.y = v[(size_t)(j0 + kk + 1) * D_MODEL + head * HDIM + l16];
      o = wmma4(a, b, o);
      if (INTRA) {
        v2f bx;
        bx.x = (l16 < 3) ? xin[(j0 + kk) * 3 + l16] : 0.f;
        bx.y = (l16 < 3) ? xin[(j0 + kk + 1) * 3 + l16] : 0.f;
        xa = wmma4(a, bx, xa);
      }
    }
  }

  #pragma unroll
  for (int r = 0; r < 8; ++r) {
    const int row = i0 + r + half * 8;
    const float invl = 1.0f / l_[r];
    out[(size_t)row * D_MODEL + head * HDIM + l16] = o[r] * invl;
    if (INTRA && l16 < 3)
      atomicAdd(&px[row * 3 + l16], xa[r] * invl * (1.0f / NHEAD));
  }
}

// ---------------------------------------------------------------------------
// Coordinate update: gate = tanh(h@Wc + bc); x += gate*(x - px). One wave/edge.
// ---------------------------------------------------------------------------
__global__ __launch_bounds__(256) void coord_k(const float* __restrict__ h,
                                               const float* __restrict__ Wc,
                                               const float* __restrict__ bc,
                                               const float* __restrict__ px,
                                               float* __restrict__ x) {
  const int lane = threadIdx.x & 31;
  const int e = (blockIdx.x * blockDim.x + threadIdx.x) >> 5;
  const float* hp = h + (size_t)e * D_MODEL;
  float s = 0.f;
  #pragma unroll
  for (int t = 0; t < 4; ++t) {
    int c = lane + 32 * t;
    s += hp[c] * Wc[c];
  }
  #pragma unroll
  for (int off = 16; off; off >>= 1) s += __shfl_xor(s, off, 32);
  const float g = tanhf(s + bc[0]);
  if (lane < 3) {
    float xv = x[e * 3 + lane];
    x[e * 3 + lane] = xv + g * (xv - px[e * 3 + lane]);
  }
}

// ---------------------------------------------------------------------------
// host orchestration
// ---------------------------------------------------------------------------
extern "C" void kernel_launch(void* const* d_in, const int* in_sizes, int n_in,
                              void* d_out, int out_size, void* d_ws, size_t ws_size,
                              hipStream_t stream) {
  (void)in_sizes; (void)n_in; (void)out_size; (void)ws_size;
  const int E = E_EDGES;

  const float* ef   = (const float*)d_in[0];
  const float* ec   = (const float*)d_in[1];
  const int*   ei   = (const int*)  d_in[2];
  const float* ncp  = (const float*)d_in[3];
  const int*   bidp = (const int*)  d_in[4];
  const float* Wq_a = (const float*)d_in[5];
  const float* Wk_a = (const float*)d_in[6];
  const float* Wv_a = (const float*)d_in[7];
  const float* Wo_a = (const float*)d_in[8];
  const float* Wq_e = (const float*)d_in[9];
  const float* Wk_e = (const float*)d_in[10];
  const float* Wv_e = (const float*)d_in[11];
  const float* Wo_e = (const float*)d_in[12];
  const float* W1   = (const float*)d_in[13];
  const float* b1   = (const float*)d_in[14];
  const float* W2   = (const float*)d_in[15];
  const float* b2   = (const float*)d_in[16];
  const float* ln1g = (const float*)d_in[17];
  const float* ln1b = (const float*)d_in[18];
  const float* ln2g = (const float*)d_in[19];
  const float* ln2b = (const float*)d_in[20];
  const float* ln3g = (const float*)d_in[21];
  const float* ln3b = (const float*)d_in[22];
  const float* Wrbf = (const float*)d_in[23];
  const float* Wc   = (const float*)d_in[24];
  const float* bc   = (const float*)d_in[25];
  float* out = (float*)d_out;

  // workspace carve-out
  char* w = (char*)d_ws;
  auto carve = [&](size_t bytes) -> char* {
    char* p = w;
    w += (bytes + 255) & ~(size_t)255;
    return p;
  };
  float*    h    = (float*)carve((size_t)E * 128 * 4);
  float*    qb   = (float*)carve((size_t)E * 128 * 4);
  float*    kb   = (float*)carve((size_t)E * 128 * 4);
  float*    vb   = (float*)carve((size_t)E * 128 * 4);
  float*    ao   = (float*)carve((size_t)E * 128 * 4);
  float*    tb   = (float*)carve((size_t)E * 128 * 4);
  float*    f1   = (float*)carve((size_t)E * FFN_D * 4);
  float*    rbfb = (float*)carve((size_t)E * RBF_R * 4);
  float*    px   = (float*)carve((size_t)E * 3 * 4);
  float*    xb   = (float*)carve((size_t)E * 3 * 4);
  unsigned* intraM = (unsigned*)carve((size_t)E * EW * 4);
  unsigned* interM = (unsigned*)carve((size_t)E * EW * 4);

  const int TPB = 256;
  // one wave per 16x32 tile, 8 waves per block
  const int g_tiles128 = (E / 16) * (128 / 32) / 8;  // 96 blocks
  const int g_tiles512 = (E / 16) * (512 / 32) / 8;  // 384 blocks
  const int g_attn     = (E / 16) * NHEAD / 8;       // 192 blocks
  const int g_rows     = E * 32 / TPB;               // one wave per row

  copy_f32<<<(E * 128 + TPB - 1) / TPB, TPB, 0, stream>>>(ef, h, E * 128);
  copy_f32<<<(E * 3 + TPB - 1) / TPB, TPB, 0, stream>>>(ec, xb, E * 3);
  rbf_kernel<<<(E + TPB - 1) / TPB, TPB, 0, stream>>>(ei, ncp, rbfb);
  mask_kernel<<<(E * EW + TPB - 1) / TPB, TPB, 0, stream>>>(ei, bidp, intraM, interM);

  for (int l = 0; l < 3; ++l) {
    const float* wq_a = Wq_a + (size_t)l * 128 * 128;
    const float* wk_a = Wk_a + (size_t)l * 128 * 128;
    const float* wv_a = Wv_a + (size_t)l * 128 * 128;
    const float* wo_a = Wo_a + (size_t)l * 128 * 128;
    const float* wq_e = Wq_e + (size_t)l * 128 * 128;
    const float* wk_e = Wk_e + (size_t)l * 128 * 128;
    const float* wv_e = Wv_e + (size_t)l * 128 * 128;
    const float* wo_e = Wo_e + (size_t)l * 128 * 128;

    // h = h + rbf @ Wrbf[l]
    gemm_wmma<0, true, false><<<g_tiles128, TPB, 0, stream>>>(
        rbfb, Wrbf + (size_t)l * RBF_R * 128, nullptr, h, h, E, 128, RBF_R);

    // intra-block attention
    gemm_wmma<0, false, false><<<g_tiles128, TPB, 0, stream>>>(h, wq_a, nullptr, nullptr, qb, E, 128, 128);
    gemm_wmma<0, false, false><<<g_tiles128, TPB, 0, stream>>>(h, wk_a, nullptr, nullptr, kb, E, 128, 128);
    gemm_wmma<0, false, false><<<g_tiles128, TPB, 0, stream>>>(h, wv_a, nullptr, nullptr, vb, E, 128, 128);
    zero_f32<<<(E * 3 + TPB - 1) / TPB, TPB, 0, stream>>>(px, E * 3);
    attn_flash<true><<<g_attn, TPB, 0, stream>>>(qb, kb, vb, intraM, xb, ao, px);
    gemm_wmma<0, true, false><<<g_tiles128, TPB, 0, stream>>>(ao, wo_a, nullptr, h, tb, E, 128, 128);
    layernorm_k<<<g_rows, TPB, 0, stream>>>(tb, ln1g + l * 128, ln1b + l * 128, h);

    // inter-block attention
    gemm_wmma<0, false, false><<<g_tiles128, TPB, 0, stream>>>(h, wq_e, nullptr, nullptr, qb, E, 128, 128);
    gemm_wmma<0, false, false><<<g_tiles128, TPB, 0, stream>>>(h, wk_e, nullptr, nullptr, kb, E, 128, 128);
    gemm_wmma<0, false, false><<<g_tiles128, TPB, 0, stream>>>(h, wv_e, nullptr, nullptr, vb, E, 128, 128);
    attn_flash<false><<<g_attn, TPB, 0, stream>>>(qb, kb, vb, interM, nullptr, ao, nullptr);
    gemm_wmma<0, true, false><<<g_tiles128, TPB, 0, stream>>>(ao, wo_e, nullptr, h, tb, E, 128, 128);
    layernorm_k<<<g_rows, TPB, 0, stream>>>(tb, ln2g + l * 128, ln2b + l * 128, h);

    // FFN
    gemm_wmma<1, false, true><<<g_tiles512, TPB, 0, stream>>>(
        h, W1 + (size_t)l * 128 * FFN_D, b1 + (size_t)l * FFN_D, nullptr, f1, E, FFN_D, 128);
    gemm_wmma<0, true, true><<<g_tiles128, TPB, 0, stream>>>(
        f1, W2 + (size_t)l * FFN_D * 128, b2 + (size_t)l * 128, h, tb, E, 128, FFN_D);
    layernorm_k<<<g_rows, TPB, 0, stream>>>(tb, ln3g + l * 128, ln3b + l * 128, h);

    // coordinate update
    coord_k<<<g_rows, TPB, 0, stream>>>(h, Wc + l * 128, bc + l, px, xb);
  }

  copy_f32<<<(E * 128 + TPB - 1) / TPB, TPB, 0, stream>>>(h, out, E * 128);
  copy_f32<<<(E * 3 + TPB - 1) / TPB, TPB, 0, stream>>>(xb, out + (size_t)E * 128, E * 3);
}